// NeuralAESImplementation_34883724378317
// MI455X (gfx1250) — compile-verified
//
#include <hip/hip_runtime.h>

typedef float v2f __attribute__((ext_vector_type(2)));
typedef float v4f __attribute__((ext_vector_type(4)));
typedef float v8f __attribute__((ext_vector_type(8)));

#define NW 8          // waves per block (256 threads, wave32)
#define G  2          // AES states (batch items) per wave
#define NROUNDS 10

// wave-local LDS ordering: DS ops from this wave complete before we proceed;
// asm memory clobber stops the compiler reordering LDS accesses across it.
__device__ __forceinline__ void lds_wait() {
  asm volatile("s_wait_dscnt 0" ::: "memory");
}

// f32 WMMA: D[16x16] = A[16x4] * B[4x16] + C   (exact fp32 math)
__device__ __forceinline__ v8f wmma4(v2f a, v2f b, v8f c) {
  return __builtin_amdgcn_wmma_f32_16x16x4_f32(false, a, false, b, (short)0, c,
                                               false, false);
}

// relu(w00*s + w01*k + b0) + relu(w10*s + w11*k + b1)  (== |s-k| for c=1)
__device__ __forceinline__ float xor_gadget(float s, float k,
                                            float w00, float w01, float w10,
                                            float w11, float b0, float b1) {
  float c0 = fmaf(w00, s, fmaf(w01, k, b0));
  float c1 = fmaf(w10, s, fmaf(w11, k, b1));
  return fmaxf(c0, 0.0f) + fmaxf(c1, 0.0f);
}

// relu(s) - 2 relu(s-1) + 2 relu(s-2) - 2 relu(s-3)  (parity of XOR-of-3)
__device__ __forceinline__ float parity_gadget(float s) {
  return fmaxf(s, 0.0f) - 2.0f * fmaxf(s - 1.0f, 0.0f)
       + 2.0f * fmaxf(s - 2.0f, 0.0f) - 2.0f * fmaxf(s - 3.0f, 0.0f);
}

__global__ __launch_bounds__(NW * 32)
void neural_aes_kernel(const float* __restrict__ stateIn,   // [B,128]
                       const float* __restrict__ roundKeys, // [11,1,4,4,8]
                       const float* __restrict__ xorw,      // [2,2]
                       const float* __restrict__ xorb,      // [2]
                       const float* __restrict__ w0,        // [32,8]
                       const float* __restrict__ b0g,       // [32]
                       const float* __restrict__ w1,        // [8,32]
                       float* __restrict__ outp,            // [B,128]
                       int nItems) {
  // per-wave state buffers: [16 bytes][12 (8 bits + pad)] floats, stride 12
  // keeps v2f loads 8B-aligned and LDS bank-conflict-free across 32 lanes.
  __shared__ __align__(16) float kLDS[(NROUNDS + 1) * 128];
  __shared__ __align__(16) float stLDS[NW][G][16 * 12]; // round-input state
  __shared__ __align__(16) float shLDS[NW][G][16 * 12]; // post-shift_rows
  __shared__ __align__(16) float hLDS[NW][16 * 34];     // H^T: [byte][32h+pad]

  const int tid  = threadIdx.x;
  const int w    = tid >> 5;
  const int lane = tid & 31;
  const int n    = lane & 15; // byte slot / matrix row (M = lane%16)
  const int up   = lane >> 4; // lane-half selects K-subrange in A/B layouts
  const int c    = n >> 2;    // column index of byte n
  const int r    = n & 3;     // row index of byte n
  const int hb   = up * 4;    // which 4 bits this lane handles elementwise

  // stage all round keys in LDS once (shared, read-only afterwards)
  for (int i = tid; i < (NROUNDS + 1) * 128; i += blockDim.x)
    kLDS[i] = roundKeys[i];
  __syncthreads();

  const float w00 = xorw[0], w01 = xorw[1], w10 = xorw[2], w11 = xorw[3];
  const float xb0 = xorb[0], xb1 = xorb[1];

  // ---- preload S-box weights into WMMA A-layout registers ----
  // A 16x4 f32 layout: lane M = lane&15; lanes 0-15 hold (K0,K1) in the two
  // regs, lanes 16-31 hold (K2,K3)  ->  per lane one float2 per K-step.
  v2f aW0[2][2]; // matmul1 A: W0 rows 16t..16t+15, K-chunk 4s..4s+3
  #pragma unroll
  for (int t = 0; t < 2; ++t)
    #pragma unroll
    for (int s = 0; s < 2; ++s)
      aW0[t][s] = *(const v2f*)(w0 + (16 * t + n) * 8 + 4 * s + 2 * up);

  v2f aW1[8]; // matmul2 A: W1 padded 8->16 rows with zeros, K = 32 (8 steps)
  #pragma unroll
  for (int s = 0; s < 8; ++s) {
    v2f v = {0.0f, 0.0f};
    if (n < 8) v = *(const v2f*)(w1 + n * 32 + 4 * s + 2 * up);
    aW1[s] = v;
  }

  // bias in D-layout: D vgpr v / lane-half up holds hidden = 16t + 8*up + v
  v8f bb0v, bb1v;
  #pragma unroll
  for (int v = 0; v < 8; ++v) {
    bb0v[v] = b0g[8 * up + v];
    bb1v[v] = b0g[16 + 8 * up + v];
  }

  const int waveBase = (blockIdx.x * NW + w) * G;
  float* Hp = &hLDS[w][0];

  // ---- initial AddRoundKey: global -> LDS state ----
  // st[b][c][r][bit] = input[b][r*32 + c*8 + bit]
  #pragma unroll
  for (int g = 0; g < G; ++g) {
    int item = waveBase + g;
    if (item >= nItems) continue;
    float* stg = &stLDS[w][g][0];
    v4f x  = *(const v4f*)(stateIn + (size_t)item * 128 + r * 32 + c * 8 + hb);
    v4f kk = *(const v4f*)(&kLDS[c * 32 + r * 8 + hb]);
    v4f res;
    #pragma unroll
    for (int j = 0; j < 4; ++j)
      res[j] = xor_gadget(x[j], kk[j], w00, w01, w10, w11, xb0, xb1);
    *(v4f*)(stg + n * 12 + hb) = res;
  }
  lds_wait();

  for (int round = 1; round <= NROUNDS; ++round) {
    #pragma unroll
    for (int g = 0; g < G; ++g) {
      int item = waveBase + g;
      if (item >= nItems) continue;
      float* stg = &stLDS[w][g][0];
      float* shg = &shLDS[w][g][0];

      // ---- SubBytes matmul #1: H^T[32x16] = W0[32x8] * X^T[8x16] ----
      // B 4x16 layout: lane N = lane&15; lanes 0-15 carry (K0,K1),
      // lanes 16-31 carry (K2,K3): one 8B LDS load per K-step.
      v8f h0 = {}, h1 = {};
      {
        v2f bA = *(const v2f*)(stg + n * 12 + 0 + 2 * up); // bits 0..3
        v2f bB = *(const v2f*)(stg + n * 12 + 4 + 2 * up); // bits 4..7
        h0 = wmma4(aW0[0][0], bA, h0);
        h0 = wmma4(aW0[0][1], bB, h0);
        h1 = wmma4(aW0[1][0], bA, h1);
        h1 = wmma4(aW0[1][1], bB, h1);
      }
      // bias + relu, store H^T to LDS as [byte][hidden] (the C->B transpose)
      #pragma unroll
      for (int v = 0; v < 8; v += 2) {
        v2f p = { fmaxf(h0[v] + bb0v[v], 0.0f),
                  fmaxf(h0[v + 1] + bb0v[v + 1], 0.0f) };
        *(v2f*)(Hp + n * 34 + 8 * up + v) = p;
        v2f q = { fmaxf(h1[v] + bb1v[v], 0.0f),
                  fmaxf(h1[v + 1] + bb1v[v + 1], 0.0f) };
        *(v2f*)(Hp + n * 34 + 16 + 8 * up + v) = q;
      }
      lds_wait();

      // ---- SubBytes matmul #2: Y^T[16x16] = W1pad[16x32] * H^T[32x16] ----
      v8f y = {};
      #pragma unroll
      for (int s = 0; s < 8; ++s) {
        v2f bv = *(const v2f*)(Hp + n * 34 + 4 * s + 2 * up);
        y = wmma4(aW1[s], bv, y);
      }

      // ---- ShiftRows fused into writeback ----
      // lane n (<16) holds byte n's 8 output bits in y[0..7];
      // new column c' = (c - r) mod 4.
      if (lane < 16) {
        int np = (((c - r) & 3) << 2) | r;
        float* p = shg + np * 12;
        #pragma unroll
        for (int v = 0; v < 8; v += 2) {
          v2f pv = { y[v], y[v + 1] };
          *(v2f*)(p + v) = pv;
        }
      }
      lds_wait();
    }

    if (round < NROUNDS) {
      // ---- MixColumns (XOR-of-3 parity gadget) + AddRoundKey ----
      #pragma unroll
      for (int g = 0; g < G; ++g) {
        int item = waveBase + g;
        if (item >= nItems) continue;
        float* stg = &stLDS[w][g][0];
        const float* shg = &shLDS[w][g][0];
        v4f x0 = *(const v4f*)(shg + (c * 4 + r) * 12 + hb);
        v4f x1 = *(const v4f*)(shg + (c * 4 + ((r + 1) & 3)) * 12 + hb);
        v4f x2 = *(const v4f*)(shg + (c * 4 + ((r + 2) & 3)) * 12 + hb);
        v4f kk = *(const v4f*)(&kLDS[round * 128 + c * 32 + r * 8 + hb]);
        v4f res;
        #pragma unroll
        for (int j = 0; j < 4; ++j) {
          float s = x0[j] + x1[j] + x2[j];
          res[j] = xor_gadget(parity_gadget(s), kk[j],
                              w00, w01, w10, w11, xb0, xb1);
        }
        *(v4f*)(stg + n * 12 + hb) = res;
      }
      lds_wait();
    } else {
      // ---- final round: AddRoundKey(k10) and write straight to HBM ----
      #pragma unroll
      for (int g = 0; g < G; ++g) {
        int item = waveBase + g;
        if (item >= nItems) continue;
        const float* shg = &shLDS[w][g][0];
        v4f x  = *(const v4f*)(shg + n * 12 + hb);
        v4f kk = *(const v4f*)(&kLDS[NROUNDS * 128 + c * 32 + r * 8 + hb]);
        v4f res;
        #pragma unroll
        for (int j = 0; j < 4; ++j)
          res[j] = xor_gadget(x[j], kk[j], w00, w01, w10, w11, xb0, xb1);
        *(v4f*)(outp + (size_t)item * 128 + r * 32 + c * 8 + hb) = res;
      }
    }
  }
}

extern "C" void kernel_launch(void* const* d_in, const int* in_sizes, int n_in,
                              void* d_out, int out_size, void* d_ws,
                              size_t ws_size, hipStream_t stream) {
  (void)n_in; (void)out_size; (void)d_ws; (void)ws_size;
  const float* stateIn   = (const float*)d_in[0];
  const float* roundKeys = (const float*)d_in[1];
  const float* xorwp     = (const float*)d_in[2];
  const float* xorbp     = (const float*)d_in[3];
  const float* w0        = (const float*)d_in[4];
  const float* b0        = (const float*)d_in[5];
  const float* w1        = (const float*)d_in[6];
  float* out             = (float*)d_out;

  int nItems = in_sizes[0] / 128;                 // 131072
  int itemsPerBlock = NW * G;                     // 16
  int blocks = (nItems + itemsPerBlock - 1) / itemsPerBlock; // 8192

  neural_aes_kernel<<<blocks, NW * 32, 0, stream>>>(
      stateIn, roundKeys, xorwp, xorbp, w0, b0, w1, out, nItems);
}